// EventEmbeddingModel_7421703488265
// MI455X (gfx1250) — compile-verified
//
#include <hip/hip_runtime.h>

// Problem constants from the reference
#define BATCH   8192
#define HLEN    64
#define DIM     256
#define ROWS    16          // batch rows per block == WMMA M tile
#define HSTRIDE 260         // his row stride in LDS (floats): 16B-aligned rows, 64-bank conflict-free

typedef __attribute__((ext_vector_type(2))) float v2f;
typedef __attribute__((ext_vector_type(4))) float v4f;
typedef __attribute__((ext_vector_type(8))) float v8f;

__global__ __launch_bounds__(256) void event_emb_fused(
    const int*   __restrict__ ent_ids,     // [B]
    const float* __restrict__ cur_time,    // [B]
    const int*   __restrict__ hist_ids,    // [B, L]
    const float* __restrict__ hist_times,  // [B, L]
    const int*   __restrict__ hist_len,    // [B]
    const float* __restrict__ emb,         // [V, D]
    const float* __restrict__ Wm,          // [D, D]  (out = his @ W^T + b)
    const float* __restrict__ bias,        // [D]
    float*       __restrict__ out)         // [B, D]
{
    __shared__ float s_w  [ROWS * HLEN];       // decay weights
    __shared__ int   s_id [ROWS * HLEN];       // history ids
    __shared__ float s_his[ROWS * HSTRIDE];    // weighted-sum result (A matrix)

    const int t       = threadIdx.x;
    const int rowBase = blockIdx.x * ROWS;

    // ---------- Phase 0: weights + ids into LDS (one exp per (r,l)) ----------
    #pragma unroll
    for (int i = 0; i < 4; ++i) {
        const int idx = t + 256 * i;           // 0..1023 covers 16x64
        const int r   = idx >> 6;
        const int l   = idx & (HLEN - 1);
        const int row = rowBase + r;
        const int len = hist_len[row];
        const float ct = cur_time[row];
        const float tt = hist_times[(size_t)row * HLEN + l];
        s_id[idx] = hist_ids[(size_t)row * HLEN + l];
        s_w[idx]  = (l < len) ? __expf(tt - ct) : 0.0f;
    }
    __syncthreads();

    // ---------- Phase 1: ragged weighted gather-sum (memory bound) ----------
    // 64 threads per row, 4 floats per thread -> global_load_b128 per lane.
    // rSub = t>>6 is wave-uniform, so the len loop bound stays scalar.
    {
        const int rSub = t >> 6;               // 0..3 (wave-uniform)
        const int d4   = (t & 63) * 4;         // column quad
        #pragma unroll
        for (int rr = 0; rr < 4; ++rr) {
            const int r   = rr * 4 + rSub;
            const int row = rowBase + r;
            const int len = hist_len[row];     // uniform within wave
            v4f acc = {0.0f, 0.0f, 0.0f, 0.0f};
            if (len > 0) {
                const int base = r * HLEN;
                for (int l = 0; l < len; ++l) {
                    const int   id = s_id[base + l];
                    const float w  = s_w [base + l];
                    const v4f e = *(const v4f*)(emb + (size_t)id * DIM + d4); // b128
                    acc += w * e;
                }
            } else {
                acc = *(const v4f*)(emb + (size_t)ent_ids[row] * DIM + d4);
            }
            *(v4f*)(&s_his[r * HSTRIDE + d4]) = acc;   // ds_store_b128 (16B aligned)
        }
    }
    __syncthreads();

    // ---------- Phase 2: out[16,256] = his[16,256] @ W^T + bias via WMMA ----
    // Wave w handles N-tiles {w, w+8}; K-loop in steps of 4 (f32 WMMA).
    const int lane = t & 31;
    const int wave = t >> 5;
    const int hi   = lane >> 4;        // 0: K pair {0,1} / M 0..7 ; 1: K {2,3} / M 8..15
    const int lm   = lane & 15;        // M for A, N for B/C/D

    const int col0 = wave * 16 + lm;           // N-tile 0 column
    const int col1 = col0 + 128;               // N-tile 1 column

    v8f c0, c1;
    {
        const float b0 = bias[col0];
        const float b1 = bias[col1];
        #pragma unroll
        for (int i = 0; i < 8; ++i) { c0[i] = b0; c1[i] = b1; }
    }

    // B operand: B[k][n] = W^T[k][n] = W[n][k]; lane needs W[col][k0+2*hi .. +1]
    const float* w0 = Wm + (size_t)col0 * DIM + 2 * hi;
    const float* w1 = Wm + (size_t)col1 * DIM + 2 * hi;
    // A operand: his[lm][k0+2*hi .. +1] from LDS
    const float* ar = &s_his[lm * HSTRIDE + 2 * hi];

    #pragma unroll 8
    for (int k0 = 0; k0 < DIM; k0 += 4) {
        v2f a;  a[0]  = ar[k0]; a[1]  = ar[k0 + 1];     // ds_load_2addr_b64
        v2f b0; b0[0] = w0[k0]; b0[1] = w0[k0 + 1];     // global_load_b64
        v2f b1; b1[0] = w1[k0]; b1[1] = w1[k0 + 1];
        c0 = __builtin_amdgcn_wmma_f32_16x16x4_f32(
                 false, a, false, b0, (short)0, c0, false, false);
        c1 = __builtin_amdgcn_wmma_f32_16x16x4_f32(
                 false, a, false, b1, (short)0, c1, false, false);
    }

    // D layout: VGPR v -> M = v + 8*hi, N = lm
    #pragma unroll
    for (int v = 0; v < 8; ++v) {
        const int m = v + 8 * hi;
        out[(size_t)(rowBase + m) * DIM + col0] = c0[v];
        out[(size_t)(rowBase + m) * DIM + col1] = c1[v];
    }
}

extern "C" void kernel_launch(void* const* d_in, const int* in_sizes, int n_in,
                              void* d_out, int out_size, void* d_ws, size_t ws_size,
                              hipStream_t stream) {
    (void)in_sizes; (void)n_in; (void)out_size; (void)d_ws; (void)ws_size;
    const int*   ent_ids    = (const int*)  d_in[0];
    const float* cur_time   = (const float*)d_in[1];
    const int*   hist_ids   = (const int*)  d_in[2];
    const float* hist_times = (const float*)d_in[3];
    const int*   hist_len   = (const int*)  d_in[4];
    const float* emb        = (const float*)d_in[5];
    const float* Wm         = (const float*)d_in[6];
    const float* bias       = (const float*)d_in[7];
    float* out = (float*)d_out;

    dim3 grid(BATCH / ROWS);   // 512 blocks
    dim3 block(256);           // 8 wave32
    hipLaunchKernelGGL(event_emb_fused, grid, block, 0, stream,
                       ent_ids, cur_time, hist_ids, hist_times, hist_len,
                       emb, Wm, bias, out);
}